// GraphGather_498216207038
// MI455X (gfx1250) — compile-verified
//
#include <hip/hip_runtime.h>
#include <math.h>

// ---------------------------------------------------------------------------
// GraphGather (segment mean + segment max, sorted membership) for MI455X.
//
// Memory-bound: 1 GiB feature stream @ 23.3 TB/s -> ~46 us floor.
// One workgroup (4 wave32) per molecule; uniform binary search gives the
// contiguous atom range. Column sums run on the CDNA5 matrix path with
// V_WMMA_F32_16X16X4_F32 (A = all-ones 16x4 f32, B = 4 atoms x 16 features,
// exact f32 math). Max is fused on the same loaded registers (single pass).
//
// Hot loop is branch-free: tail atoms are peeled off, loads are unguarded
// with compile-time immediate offsets off a single incremented base pointer,
// and 4 independent WMMA accumulator chains hide the C->C accumulate latency.
// ---------------------------------------------------------------------------

typedef __attribute__((ext_vector_type(2))) float v2f;
typedef __attribute__((ext_vector_type(8))) float v8f;

#define DFEAT 128

__global__ __launch_bounds__(128) void graph_gather_wmma_kernel(
    const float* __restrict__ feat,   // [n_atoms, 128]
    const int*   __restrict__ mem,    // [n_atoms], sorted
    float*       __restrict__ out,    // [batch, 256] = [mean(128) | max(128)]
    int n_atoms)
{
  const int mol = blockIdx.x;

  // ---- uniform binary search: first index with mem[i] >= mol, and >= mol+1
  int lo = 0, hi = n_atoms;
  while (lo < hi) { int mid = (lo + hi) >> 1; if (mem[mid] <  mol) lo = mid + 1; else hi = mid; }
  const int start = lo;
  hi = n_atoms;
  while (lo < hi) { int mid = (lo + hi) >> 1; if (mem[mid] <= mol) lo = mid + 1; else hi = mid; }
  const int end = lo;
  const int n   = end - start;

  const int lane = threadIdx.x & 31;
  const int wave = threadIdx.x >> 5;   // 0..3, owns features [32*wave, 32*wave+32)
  const int col  = lane & 15;          // N position within a 16-wide tile
  const int half = lane >> 4;          // 0: lanes 0-15, 1: lanes 16-31

  const int fb0 = wave * 32;           // first 16-feature tile
  const int fb1 = fb0 + 16;            // second 16-feature tile

  // A = all-ones 16x4 f32 matrix (2 VGPRs, every lane holds {1,1}).
  v2f ones; ones.x = 1.0f; ones.y = 1.0f;

  v8f c0a = {}, c0b = {};              // independent f32 sum chains, tile 0
  v8f c1a = {}, c1b = {};              // independent f32 sum chains, tile 1
  float m0 = -INFINITY, m1 = -INFINITY;

  // ---- branch-free main loop: 8 atoms / iteration, unconditional NT loads.
  // B layout (4x16 f32, 2 VGPRs): lanes 0-15 hold one K-row, lanes 16-31
  // another. K ordering is irrelevant (A all-ones -> D = sum over K of B).
  const int n8 = n & ~7;
  {
    // lane's base: row (start + half), feature fb0 + col.
    const float* p = feat + (size_t)(start + half) * DFEAT + (fb0 + col);
    for (int a = 0; a < n8; a += 8) {
      // group 0: atoms a+0..a+3        group 1: atoms a+4..a+7
      const float t0a = __builtin_nontemporal_load(p + 0);          // rows +0/+1, tile0
      const float t1a = __builtin_nontemporal_load(p + 16);         // rows +0/+1, tile1
      const float t0b = __builtin_nontemporal_load(p + 2 * DFEAT);        // rows +2/+3
      const float t1b = __builtin_nontemporal_load(p + 2 * DFEAT + 16);
      const float u0a = __builtin_nontemporal_load(p + 4 * DFEAT);        // rows +4/+5
      const float u1a = __builtin_nontemporal_load(p + 4 * DFEAT + 16);
      const float u0b = __builtin_nontemporal_load(p + 6 * DFEAT);        // rows +6/+7
      const float u1b = __builtin_nontemporal_load(p + 6 * DFEAT + 16);
      p += 8 * DFEAT;

      v2f b0; b0.x = t0a; b0.y = t0b;
      v2f b1; b1.x = t1a; b1.y = t1b;
      v2f d0; d0.x = u0a; d0.y = u0b;
      v2f d1; d1.x = u1a; d1.y = u1b;

      c0a = __builtin_amdgcn_wmma_f32_16x16x4_f32(false, ones, false, b0,
                                                  (short)0, c0a, false, false);
      c1a = __builtin_amdgcn_wmma_f32_16x16x4_f32(false, ones, false, b1,
                                                  (short)0, c1a, false, false);
      c0b = __builtin_amdgcn_wmma_f32_16x16x4_f32(false, ones, false, d0,
                                                  (short)0, c0b, false, false);
      c1b = __builtin_amdgcn_wmma_f32_16x16x4_f32(false, ones, false, d1,
                                                  (short)0, c1b, false, false);

      m0 = fmaxf(fmaxf(m0, t0a), fmaxf(t0b, fmaxf(u0a, u0b)));
      m1 = fmaxf(fmaxf(m1, t1a), fmaxf(t1b, fmaxf(u1a, u1b)));
    }
  }

  // ---- scalar tail (<= 7 atoms), outside the hot path.
  float e0 = 0.0f, e1 = 0.0f;
  if (half == 0) {
    for (int a = start + n8; a < end; ++a) {
      const float* q = feat + (size_t)a * DFEAT + col;
      const float x0 = __builtin_nontemporal_load(q + fb0);
      const float x1 = __builtin_nontemporal_load(q + fb1);
      e0 += x0;  e1 += x1;
      m0 = fmaxf(m0, x0);
      m1 = fmaxf(m1, x1);
    }
  }

  // Merge the two half-wave partial maxima (wave32 xor-16 swap). Tail maxima
  // live only on half 0 lanes, which are also the writers -> still correct.
  m0 = fmaxf(m0, __shfl_xor(m0, 16, 32));
  m1 = fmaxf(m1, __shfl_xor(m1, 16, 32));

  // C/D layout: VGPR0, lanes 0-15 = row M=0, N=lane. All D rows identical
  // (A all-ones), so c[0] is the column sum on every lane of a half.
  const float cnt = (float)n;
  if (half == 0) {
    const float s0 = c0a[0] + c0b[0] + e0;
    const float s1 = c1a[0] + c1b[0] + e1;
    float* orow = out + (size_t)mol * (2 * DFEAT);
    orow[fb0 + col]         = s0 / cnt;      // mean (0/0 -> NaN, as reference)
    orow[fb1 + col]         = s1 / cnt;
    orow[DFEAT + fb0 + col] = m0;            // max (-inf for empty segment)
    orow[DFEAT + fb1 + col] = m1;
  }
}

extern "C" void kernel_launch(void* const* d_in, const int* in_sizes, int n_in,
                              void* d_out, int out_size, void* d_ws, size_t ws_size,
                              hipStream_t stream) {
  const float* feat = (const float*)d_in[0];     // atom_features [N_ATOMS*128] f32
  const int*   mem  = (const int*)d_in[1];       // membership   [N_ATOMS]     i32
  const int n_atoms = in_sizes[1];               // flat count of membership
  const int batch   = out_size / (2 * DFEAT);    // [batch, 256] output

  graph_gather_wmma_kernel<<<batch, 128, 0, stream>>>(feat, mem, (float*)d_out,
                                                      n_atoms);
}